// RNN_GRU_1735166787668
// MI455X (gfx1250) — compile-verified
//
#include <hip/hip_runtime.h>

#define E_   10
#define H_   128
#define H3_  384
#define O_   21
#define L_   32768

typedef __attribute__((ext_vector_type(2))) float v2f;
typedef __attribute__((ext_vector_type(8))) float v8f;

// ---------------------------------------------------------------------------
// Phase 1: x_proj[t][j] = b_ih[j] + sum_e W_ih[j][e] * emb[tokens[t]][e]
// K=10 -> plain VALU FMA (WMMA K-padding would waste the tensor pipe).
// ---------------------------------------------------------------------------
__global__ void __launch_bounds__(H3_) xproj_kernel(
    const int* __restrict__ tokens, const float* __restrict__ emb,
    const float* __restrict__ W_ih, const float* __restrict__ b_ih,
    float* __restrict__ x_proj) {
  const int j = threadIdx.x;                   // 0..383
  float wrow[E_];
#pragma unroll
  for (int e = 0; e < E_; ++e) wrow[e] = W_ih[j * E_ + e];
  const float bj = b_ih[j];
  const int t0 = blockIdx.x * 16;
  for (int tt = 0; tt < 16; ++tt) {
    const int t = t0 + tt;
    const int tok = tokens[t];
    const float* ev = emb + tok * E_;
    float acc = bj;
#pragma unroll
    for (int e = 0; e < E_; ++e) acc += wrow[e] * ev[e];
    x_proj[(size_t)t * H3_ + j] = acc;
  }
}

// ---------------------------------------------------------------------------
// Phase 2: sequential GRU scan. One workgroup, 384 lanes = 12 waves.
// Lane j keeps W_hh row j (128 f32) in VGPRs. h broadcast from LDS (b128).
// Two barriers per step.
// ---------------------------------------------------------------------------
__global__ void __launch_bounds__(H3_, 1) gru_scan_kernel(
    const float* __restrict__ W_hh, const float* __restrict__ b_hh,
    const float* __restrict__ x_proj, float* __restrict__ hiddens,
    float* __restrict__ last_h) {
  __shared__ __align__(16) float lds_h[H_];
  __shared__ float lds_r[H_];
  __shared__ float lds_z[H_];
  __shared__ float lds_dn[H_];
  __shared__ float lds_xn[H_];

  const int j = threadIdx.x;
  const int grp = j >> 7;     // 0:r 1:z 2:n
  const int i = j & (H_ - 1);

  // One-time: own W_hh row into VGPRs (fully unrolled -> register array)
  float w[H_];
  const float* wr = W_hh + j * H_;
#pragma unroll
  for (int k = 0; k < H_ / 4; ++k) {
    const float4 v = ((const float4*)wr)[k];
    w[4 * k + 0] = v.x; w[4 * k + 1] = v.y;
    w[4 * k + 2] = v.z; w[4 * k + 3] = v.w;
  }
  const float bj = b_hh[j];
  if (j < H_) lds_h[j] = 0.0f;
  __syncthreads();

  for (int t = 0; t < L_; ++t) {
    const float xp = x_proj[(size_t)t * H3_ + j];
    if (t + 1 < L_)
      __builtin_prefetch(x_proj + (size_t)(t + 1) * H3_ + j, 0, 0);

    float dot = bj;
#pragma unroll
    for (int k = 0; k < H_ / 4; ++k) {
      const float4 hv = ((const float4*)lds_h)[k];   // LDS broadcast read
      dot += w[4 * k + 0] * hv.x + w[4 * k + 1] * hv.y +
             w[4 * k + 2] * hv.z + w[4 * k + 3] * hv.w;
    }

    float r = 0.0f;
    if (grp == 0) {
      r = 1.0f / (1.0f + __expf(-(xp + dot)));
      lds_r[i] = r;
    } else if (grp == 1) {
      lds_z[i] = 1.0f / (1.0f + __expf(-(xp + dot)));
    } else {
      lds_dn[i] = dot;
      lds_xn[i] = xp;
    }
    __syncthreads();

    if (grp == 0) {
      const float n = tanhf(lds_xn[i] + r * lds_dn[i]);
      const float z = lds_z[i];
      const float hn = (1.0f - z) * n + z * lds_h[i];
      lds_h[i] = hn;
      hiddens[(size_t)t * H_ + i] = hn;
    }
    __syncthreads();
  }
  if (j < H_) last_h[j] = lds_h[j];
}

// ---------------------------------------------------------------------------
// Phase 3: logits = hiddens @ W_dec^T + b_dec, then row log_softmax.
// V_WMMA_F32_16X16X4_F32: one wave per 16-row tile, 32 k-steps x 2 N-tiles.
// ---------------------------------------------------------------------------
__global__ void __launch_bounds__(256) decoder_kernel(
    const float* __restrict__ hid, const float* __restrict__ W_dec,
    const float* __restrict__ b_dec, float* __restrict__ out) {
  __shared__ __align__(16) float Wp[32 * H_];      // W_dec zero-padded to 32 rows
  for (int idx = threadIdx.x; idx < 32 * H_; idx += 256) {
    const int n = idx / H_, k = idx % H_;
    Wp[idx] = (n < O_) ? W_dec[n * H_ + k] : 0.0f;
  }
  __syncthreads();

  const int wave = threadIdx.x >> 5;
  const int lane = threadIdx.x & 31;
  const int half = lane >> 4;        // ISA: lanes 16-31 carry K+2 / M+8 halves
  const int lm = lane & 15;
  const int m0 = (blockIdx.x * 8 + wave) * 16;
  const float* arow = hid + (size_t)(m0 + lm) * H_;

  v8f c0 = {};
  v8f c1 = {};
#pragma unroll 4
  for (int kk = 0; kk < 32; ++kk) {
    const int k = kk * 4 + half * 2;
    v2f a;  a.x  = arow[k];               a.y  = arow[k + 1];
    v2f b0; b0.x = Wp[lm * H_ + k];       b0.y = Wp[lm * H_ + k + 1];
    v2f b1; b1.x = Wp[(16 + lm) * H_ + k]; b1.y = Wp[(16 + lm) * H_ + k + 1];
    c0 = __builtin_amdgcn_wmma_f32_16x16x4_f32(false, a, false, b0,
                                               (short)0, c0, false, false);
    c1 = __builtin_amdgcn_wmma_f32_16x16x4_f32(false, a, false, b1,
                                               (short)0, c1, false, false);
  }

  const float bia0 = b_dec[lm];
  const float bia1 = (lm < O_ - 16) ? b_dec[16 + lm] : 0.0f;
#pragma unroll
  for (int p = 0; p < 8; ++p) {
    const int row = m0 + half * 8 + p;           // C layout: M = half*8 + vgpr
    const float v0 = c0[p] + bia0;               // cols 0..15 at lane lm
    const float v1 = (lm < O_ - 16) ? (c1[p] + bia1) : -3.0e38f;
    float mx = fmaxf(v0, v1);
#pragma unroll
    for (int off = 8; off >= 1; off >>= 1)
      mx = fmaxf(mx, __shfl_xor(mx, off, 16));   // reduce within half-wave
    float s = __expf(v0 - mx) + ((lm < O_ - 16) ? __expf(v1 - mx) : 0.0f);
#pragma unroll
    for (int off = 8; off >= 1; off >>= 1)
      s += __shfl_xor(s, off, 16);
    const float lse = mx + __logf(s);
    out[(size_t)row * O_ + lm] = v0 - lse;
    if (lm < O_ - 16) out[(size_t)row * O_ + 16 + lm] = v1 - lse;
  }
}

// ---------------------------------------------------------------------------
extern "C" void kernel_launch(void* const* d_in, const int* in_sizes, int n_in,
                              void* d_out, int out_size, void* d_ws, size_t ws_size,
                              hipStream_t stream) {
  const int*   tokens = (const int*)d_in[0];
  const float* emb    = (const float*)d_in[1];
  const float* W_ih   = (const float*)d_in[2];
  const float* W_hh   = (const float*)d_in[3];
  const float* b_ih   = (const float*)d_in[4];
  const float* b_hh   = (const float*)d_in[5];
  const float* W_dec  = (const float*)d_in[6];
  const float* b_dec  = (const float*)d_in[7];
  float* out = (float*)d_out;

  float* x_proj  = (float*)d_ws;                      // L*384 f32 = 50.3 MB
  float* hiddens = x_proj + (size_t)L_ * H3_;         // L*128 f32 = 16.8 MB

  xproj_kernel<<<L_ / 16, H3_, 0, stream>>>(tokens, emb, W_ih, b_ih, x_proj);
  gru_scan_kernel<<<1, H3_, 0, stream>>>(W_hh, b_hh, x_proj, hiddens,
                                         out + (size_t)L_ * O_);
  decoder_kernel<<<L_ / 128, 256, 0, stream>>>(hiddens, W_dec, b_dec, out);
}